// ComplexGaussianSplatter_7962869366960
// MI455X (gfx1250) — compile-verified
//
#include <hip/hip_runtime.h>
#include <stdint.h>

// ComplexGaussianSplatter for MI455X (gfx1250) — WMMA + async-LDS version.
//
// mah(m,n) and r2(m,n) are bilinear forms in a shared 12-dim feature basis
//   f(q) = [qx2,qy2,qz2, qxqy,qxqz,qyqz, qx,qy,qz, 1, 0, 0]
// so the pre-transcendental work is two GEMMs ([M x 12] x [12 x N]) done as
// chains of 3x V_WMMA_F32_16X16X4_F32 per 16x16 tile. Using the same A for
// both GEMMs makes mah and r2 of one (m,n) land in the same lane/VGPR slot,
// so the transcendental stage (v_exp/v_sqrt/v_cos/v_sin) needs no cross-lane
// traffic. Complex field accumulates per (M-row, N-lane); one 4-step
// shfl_xor reduction over the 16 N-lanes at the end.

typedef float v2f __attribute__((ext_vector_type(2)));
typedef float v8f __attribute__((ext_vector_type(8)));

#define NGAUSS   1024
#define GSTRIDE  28                        // floats per gaussian record (112B)
#define LDS_FLOATS (NGAUSS * GSTRIDE)      // 28672 floats = 114688 B
#define NTILES   (NGAUSS / 16)             // 64 gaussian tiles
#define BLOCK    512                       // 16 waves x 16 query points

__global__ __launch_bounds__(256) void cgs_prep_kernel(
    const float* __restrict__ positions,   // (N,3)
    const float* __restrict__ cv_real,     // (N,)
    const float* __restrict__ cv_imag,     // (N,)
    const float* __restrict__ scales,      // (N,3)
    const float* __restrict__ rotations,   // (N,4) w,x,y,z
    float* __restrict__ ws,                // (N,28) packed records
    int N)
{
    int n = blockIdx.x * blockDim.x + threadIdx.x;
    if (n >= N) return;

    float qw = rotations[n * 4 + 0];
    float qx = rotations[n * 4 + 1];
    float qy = rotations[n * 4 + 2];
    float qz = rotations[n * 4 + 3];
    float inv = __builtin_amdgcn_rsqf(qw * qw + qx * qx + qy * qy + qz * qz);
    qw *= inv; qx *= inv; qy *= inv; qz *= inv;

    float R00 = 1.0f - 2.0f * (qy * qy + qz * qz);
    float R01 = 2.0f * (qx * qy - qw * qz);
    float R02 = 2.0f * (qx * qz + qw * qy);
    float R10 = 2.0f * (qx * qy + qw * qz);
    float R11 = 1.0f - 2.0f * (qx * qx + qz * qz);
    float R12 = 2.0f * (qy * qz - qw * qx);
    float R20 = 2.0f * (qx * qz - qw * qy);
    float R21 = 2.0f * (qy * qz + qw * qx);
    float R22 = 1.0f - 2.0f * (qx * qx + qy * qy);

    float is0 = 1.0f / scales[n * 3 + 0];
    float is1 = 1.0f / scales[n * 3 + 1];
    float is2 = 1.0f / scales[n * 3 + 2];

    // A = diag(1/s) * R^T : A[i][j] = R[j][i] * is_i
    float a00 = R00 * is0, a01 = R10 * is0, a02 = R20 * is0;
    float a10 = R01 * is1, a11 = R11 * is1, a12 = R21 * is1;
    float a20 = R02 * is2, a21 = R12 * is2, a22 = R22 * is2;

    // Q = A^T A (symmetric)
    float Q00 = a00 * a00 + a10 * a10 + a20 * a20;
    float Q11 = a01 * a01 + a11 * a11 + a21 * a21;
    float Q22 = a02 * a02 + a12 * a12 + a22 * a22;
    float Q01 = a00 * a01 + a10 * a11 + a20 * a21;
    float Q02 = a00 * a02 + a10 * a12 + a20 * a22;
    float Q12 = a01 * a02 + a11 * a12 + a21 * a22;

    float px = positions[n * 3 + 0];
    float py = positions[n * 3 + 1];
    float pz = positions[n * 3 + 2];

    // b = A p ; c = A^T b ; e = |b|^2
    float b0 = a00 * px + a01 * py + a02 * pz;
    float b1 = a10 * px + a11 * py + a12 * pz;
    float b2 = a20 * px + a21 * py + a22 * pz;
    float c0 = a00 * b0 + a10 * b1 + a20 * b2;
    float c1 = a01 * b0 + a11 * b1 + a21 * b2;
    float c2 = a02 * b0 + a12 * b1 + a22 * b2;
    float e  = b0 * b0 + b1 * b1 + b2 * b2;
    float pp = px * px + py * py + pz * pz;

    // record: [0..11]  mah coeffs  Q00,Q11,Q22,2Q01,2Q02,2Q12,-2c,-,-,e,0,0
    //         [12..23] r2  coeffs  1,1,1,0,0,0,-2p,-,-,pp,0,0
    //         [24..27] cv_re,cv_im,0,0
    float4* o = (float4*)(ws + (size_t)n * GSTRIDE);
    o[0] = make_float4(Q00, Q11, Q22, 2.0f * Q01);
    o[1] = make_float4(2.0f * Q02, 2.0f * Q12, -2.0f * c0, -2.0f * c1);
    o[2] = make_float4(-2.0f * c2, e, 0.0f, 0.0f);
    o[3] = make_float4(1.0f, 1.0f, 1.0f, 0.0f);
    o[4] = make_float4(0.0f, 0.0f, -2.0f * px, -2.0f * py);
    o[5] = make_float4(-2.0f * pz, pp, 0.0f, 0.0f);
    o[6] = make_float4(cv_real[n], cv_imag[n], 0.0f, 0.0f);
}

__global__ __launch_bounds__(BLOCK) void cgs_splat_kernel(
    const float* __restrict__ query,          // (M,3)
    const float* __restrict__ ws,             // (N,28) packed records
    const unsigned int* __restrict__ freq_p,  // scalar (low 32 bits valid)
    float* __restrict__ out,                  // (M,2) interleaved complex64
    int M)
{
    __shared__ float sh[LDS_FLOATS];          // 112 KB gaussian table

    // ---- Stage table into LDS via gfx1250 async-to-LDS path ----
    {
        unsigned tid = threadIdx.x;
        uint32_t lds_base = (uint32_t)(uintptr_t)(&sh[0]);
        unsigned long long gbase = (unsigned long long)(uintptr_t)ws;
        const unsigned nvec = (LDS_FLOATS * 4) / 16;   // 7168 x b128
#pragma unroll
        for (unsigned i = 0; i < nvec / BLOCK; ++i) {
            unsigned idx = tid + i * (unsigned)BLOCK;
            uint32_t lds_addr = lds_base + idx * 16u;
            unsigned long long gaddr = gbase + (unsigned long long)idx * 16ull;
            asm volatile("global_load_async_to_lds_b128 %0, %1, off"
                         :: "v"(lds_addr), "v"(gaddr)
                         : "memory");
        }
        asm volatile("s_wait_asynccnt 0" ::: "memory");
    }
    __syncthreads();

    unsigned tid  = threadIdx.x;
    unsigned lane = tid & 31u;
    unsigned wave = tid >> 5;                 // 0..15
    unsigned half = lane >> 4;                // 0 or 1
    unsigned mrow = lane & 15u;               // M row (for A/features), N col (for B)

    int mbase = blockIdx.x * (BLOCK / 2);     // 256 query points per block
    int mq = mbase + (int)wave * 16 + (int)mrow;
    if (mq >= M) mq = M - 1;                  // clamp, keep EXEC all-ones for WMMA

    float qx = query[mq * 3 + 0];
    float qy = query[mq * 3 + 1];
    float qz = query[mq * 3 + 2];

    // features f0..f11 of this lane's query point
    float f0 = qx * qx, f1 = qy * qy, f2 = qz * qz;
    float f3 = qx * qy, f4 = qx * qz, f5 = qy * qz;
    float f6 = qx, f7 = qy, f8 = qz, f9 = 1.0f, f10 = 0.0f, f11 = 0.0f;

    // A operands (16x4 f32): lanes 0-15 hold (K=4a,4a+1), lanes 16-31 (4a+2,4a+3)
    v2f A0, A1, A2;
    A0.x = half ? f2  : f0;  A0.y = half ? f3  : f1;
    A1.x = half ? f6  : f4;  A1.y = half ? f7  : f5;
    A2.x = half ? f10 : f8;  A2.y = half ? f11 : f9;

    const float inv_c = 1.0f / 299792458.0f;
    float k_rev = (float)freq_p[0] * inv_c;       // phase in revolutions/meter
    const float kExp = -0.72134752044448170f;     // -0.5*log2(e)

    v8f accR = {}; v8f accI = {};

    for (int t = 0; t < NTILES; ++t) {
        // per-lane gaussian g = t*16 + mrow ; B chunk a reads float2 at 4a+2*half
        const float* gb = &sh[(unsigned)(t * 16 + (int)mrow) * GSTRIDE];
        unsigned off = 2u * half;
        v2f bm0 = *(const v2f*)(gb + 0  + off);
        v2f bm1 = *(const v2f*)(gb + 4  + off);
        v2f bm2 = *(const v2f*)(gb + 8  + off);
        v2f br0 = *(const v2f*)(gb + 12 + off);
        v2f br1 = *(const v2f*)(gb + 16 + off);
        v2f br2 = *(const v2f*)(gb + 20 + off);
        v2f cv  = *(const v2f*)(gb + 24);
        float cvr = cv.x, cvi = cv.y;

        v8f mah = {};
        mah = __builtin_amdgcn_wmma_f32_16x16x4_f32(false, A0, false, bm0, (short)0, mah, false, false);
        mah = __builtin_amdgcn_wmma_f32_16x16x4_f32(false, A1, false, bm1, (short)0, mah, false, false);
        mah = __builtin_amdgcn_wmma_f32_16x16x4_f32(false, A2, false, bm2, (short)0, mah, false, false);
        v8f r2v = {};
        r2v = __builtin_amdgcn_wmma_f32_16x16x4_f32(false, A0, false, br0, (short)0, r2v, false, false);
        r2v = __builtin_amdgcn_wmma_f32_16x16x4_f32(false, A1, false, br1, (short)0, r2v, false, false);
        r2v = __builtin_amdgcn_wmma_f32_16x16x4_f32(false, A2, false, br2, (short)0, r2v, false, false);

        // elementwise: D element r is (M = r + 8*half, N = mrow) of this tile
#pragma unroll
        for (int r = 0; r < 8; ++r) {
            float amp  = __builtin_amdgcn_exp2f(mah[r] * kExp);   // v_exp_f32
            float dist = __builtin_amdgcn_sqrtf(r2v[r]);          // v_sqrt_f32
            float ph   = dist * k_rev;                            // revolutions
            float c    = __builtin_amdgcn_cosf(ph);               // v_cos_f32
            float s    = __builtin_amdgcn_sinf(ph);               // v_sin_f32
            accR[r] = fmaf(amp, fmaf(c, cvr,  s * cvi), accR[r]);
            accI[r] = fmaf(amp, fmaf(c, cvi, -s * cvr), accI[r]);
        }
    }

    // reduce over the 16 N-lanes (xor masks 1,2,4,8 stay within the half)
#pragma unroll
    for (int r = 0; r < 8; ++r) {
        float vr = accR[r], vi = accI[r];
        vr += __shfl_xor(vr, 1, 32);  vi += __shfl_xor(vi, 1, 32);
        vr += __shfl_xor(vr, 2, 32);  vi += __shfl_xor(vi, 2, 32);
        vr += __shfl_xor(vr, 4, 32);  vi += __shfl_xor(vi, 4, 32);
        vr += __shfl_xor(vr, 8, 32);  vi += __shfl_xor(vi, 8, 32);
        if (mrow == 0) {
            int mo = mbase + (int)wave * 16 + (int)half * 8 + r;
            if (mo < M) {
                out[2 * mo + 0] = vr;
                out[2 * mo + 1] = vi;
            }
        }
    }
}

extern "C" void kernel_launch(void* const* d_in, const int* in_sizes, int n_in,
                              void* d_out, int out_size, void* d_ws, size_t ws_size,
                              hipStream_t stream) {
    const float* qp  = (const float*)d_in[0];   // query_points (M,3)
    const float* pos = (const float*)d_in[1];   // positions (N,3)
    const float* cvr = (const float*)d_in[2];   // cv_real (N,)
    const float* cvi = (const float*)d_in[3];   // cv_imag (N,)
    const float* sc  = (const float*)d_in[4];   // scales (N,3)
    const float* rot = (const float*)d_in[5];   // rotations (N,4)
    const unsigned int* fq = (const unsigned int*)d_in[6]; // frequency scalar

    float* out = (float*)d_out;                 // (M,) complex64 -> 2*M floats
    float* ws  = (float*)d_ws;                  // needs N*28*4 = 114688 B scratch

    int M = in_sizes[0] / 3;
    int N = in_sizes[1] / 3;

    cgs_prep_kernel<<<(N + 255) / 256, 256, 0, stream>>>(pos, cvr, cvi, sc, rot, ws, N);

    int pts_per_block = BLOCK / 2;              // 16 waves x 16 points
    int blocks = (M + pts_per_block - 1) / pts_per_block;
    cgs_splat_kernel<<<blocks, BLOCK, 0, stream>>>(qp, ws, fq, out, M);
}